// EdgeAttention_19550691131403
// MI455X (gfx1250) — compile-verified
//
#include <hip/hip_runtime.h>
#include <hip/hip_bf16.h>

typedef __attribute__((ext_vector_type(2))) float v2f;
typedef __attribute__((ext_vector_type(8))) float v8f;

#define D_IN   128
#define TWO_D  256
#define WPAD   260   // LDS row stride (floats); 260 % 64 banks = 4 -> conflict-free across n

__global__ __launch_bounds__(256) void edge_attn_wmma_f32(
    const float* __restrict__ emb,    // [N_NODES, 128]
    const int*   __restrict__ eidx,   // [2, E]
    const float* __restrict__ W,      // [2, 256] row-major
    const float* __restrict__ bias,   // [2]
    float*       __restrict__ out,    // [E, 2]
    int E, int ntiles)
{
  __shared__ float Wlds[16 * WPAD];

  // Stage zero-padded B matrix source: Wlds[n][k] = (n<2) ? W[n][k] : 0
  for (int i = threadIdx.x; i < 16 * WPAD; i += blockDim.x) {
    int r = i / WPAD;
    int c = i - r * WPAD;
    float v = 0.0f;
    if (r < 2 && c < TWO_D) v = W[r * TWO_D + c];
    Wlds[i] = v;
  }
  __syncthreads();

  const int lane = threadIdx.x & 31;
  const int wave = threadIdx.x >> 5;
  const int n    = lane & 15;   // A-row (edge-in-tile) and B/D column
  const int h    = lane >> 4;   // half-wave selects K pair / D row-half
  const int k2   = 2 * h;

  const int tile = blockIdx.x * (blockDim.x >> 5) + wave;
  if (tile >= ntiles) return;   // uniform per wave -> EXEC stays all-ones

  // Edge feeding this lane's A rows (clamped in data only for the tail tile)
  int e  = tile * 16 + n;
  int ec = (e < E) ? e : (E - 1);
  const float* rowp = emb + (size_t)eidx[ec]     * D_IN;   // src endpoint
  const float* colp = emb + (size_t)eidx[E + ec] * D_IN;   // dst endpoint

  v8f acc0 = {}, acc1 = {}, acc2 = {}, acc3 = {};

  // K = 0..127 : src embedding half, two independent accumulator chains
  #pragma unroll 4
  for (int kb = 0; kb < D_IN; kb += 8) {
    v2f a0 = *(const v2f*)(rowp + kb + k2);
    v2f b0 = *(const v2f*)(&Wlds[n * WPAD + kb + k2]);
    acc0 = __builtin_amdgcn_wmma_f32_16x16x4_f32(false, a0, false, b0,
                                                 (short)0, acc0, false, false);
    v2f a1 = *(const v2f*)(rowp + kb + 4 + k2);
    v2f b1 = *(const v2f*)(&Wlds[n * WPAD + kb + 4 + k2]);
    acc1 = __builtin_amdgcn_wmma_f32_16x16x4_f32(false, a1, false, b1,
                                                 (short)0, acc1, false, false);
  }
  // K = 128..255 : dst embedding half
  #pragma unroll 4
  for (int kb = 0; kb < D_IN; kb += 8) {
    v2f a0 = *(const v2f*)(colp + kb + k2);
    v2f b0 = *(const v2f*)(&Wlds[n * WPAD + D_IN + kb + k2]);
    acc2 = __builtin_amdgcn_wmma_f32_16x16x4_f32(false, a0, false, b0,
                                                 (short)0, acc2, false, false);
    v2f a1 = *(const v2f*)(colp + kb + 4 + k2);
    v2f b1 = *(const v2f*)(&Wlds[n * WPAD + D_IN + kb + 4 + k2]);
    acc3 = __builtin_amdgcn_wmma_f32_16x16x4_f32(false, a1, false, b1,
                                                 (short)0, acc3, false, false);
  }

  v8f d = (acc0 + acc1) + (acc2 + acc3);

  // D layout: lane = {h,n}, VGPR v  ->  score[edge M = v + 8h][component N = n]
  float bn = bias[n & 1];

  #pragma unroll
  for (int v = 0; v < 8; ++v) {
    float s  = d[v] + bn;
    float o  = __shfl_xor(s, 1, 32);      // pair N=0 <-> N=1 (lanes 0/1, 16/17)
    float mx = fmaxf(s, o);
    float es = __expf(s - mx);
    float eo = __expf(o - mx);
    float p  = es / (es + eo);
    int eout = tile * 16 + v + 8 * h;
    if (n < 2 && eout < E) out[(size_t)eout * 2 + n] = p;
  }
}

extern "C" void kernel_launch(void* const* d_in, const int* in_sizes, int n_in,
                              void* d_out, int out_size, void* d_ws, size_t ws_size,
                              hipStream_t stream) {
  const float* emb  = (const float*)d_in[0];
  const int*   eidx = (const int*)  d_in[1];
  const float* W    = (const float*)d_in[2];
  const float* bias = (const float*)d_in[3];
  float*       out  = (float*)d_out;

  const int E      = in_sizes[1] / 2;          // edge_index is [2, E]
  const int ntiles = (E + 15) / 16;            // one 16-edge tile per wave
  const int waves_per_block = 8;               // 256 threads = 8 wave32
  const int blocks = (ntiles + waves_per_block - 1) / waves_per_block;

  edge_attn_wmma_f32<<<blocks, 256, 0, stream>>>(emb, eidx, W, bias, out, E, ntiles);
}